// Encoder_84670985274020
// MI455X (gfx1250) — compile-verified
//
#include <hip/hip_runtime.h>

// ---------------- problem constants ----------------
constexpr int B_  = 2,  L_ = 2048, D_ = 768, H_ = 12;
constexpr int E_  = 32, T_ = 8,    W_ = 4;
constexpr int R_  = 57, NN_ = 20;
constexpr int P_  = 248;            // T*(E-1)
constexpr int PPAD = 256;           // padded pair count for WMMA M tiles
constexpr int K3D = 3 * D_;         // 2304
constexpr int CB_ = 80;             // padded codebook rows (57 rel + 20 nota -> 80)

// ---------------- workspace layout (float offsets) ----------------
constexpr size_t OFF_ENT_EMB = 0;                                        // B*E*D
constexpr size_t OFF_ENT_ATT = OFF_ENT_EMB + (size_t)B_ * E_ * D_;       // B*E*H*L
constexpr size_t OFF_A       = OFF_ENT_ATT + (size_t)B_ * E_ * H_ * L_;  // B*PPAD*L
constexpr size_t OFF_C       = OFF_A       + (size_t)B_ * PPAD * L_;     // B*P*D
constexpr size_t OFF_EMBS    = OFF_C       + (size_t)B_ * P_ * D_;       // B*PPAD*3D
constexpr size_t OFF_SCORES  = OFF_EMBS    + (size_t)B_ * PPAD * K3D;    // B*PPAD*CB

typedef __attribute__((ext_vector_type(2))) float v2f;
typedef __attribute__((ext_vector_type(8))) float v8f;

// ---------------- K1a: entity embeddings (span mean-pool) ----------------
__global__ void k_ent_emb(const float* __restrict__ seq,
                          const int* __restrict__ starts,
                          float* __restrict__ ws) {
    int be = blockIdx.x;                 // [0, B*E)
    int b  = be / E_;
    int st = starts[be];
    const float* base = seq + ((size_t)b * L_ + st) * D_;
    float* out = ws + OFF_ENT_EMB + (size_t)be * D_;
    for (int d = threadIdx.x; d < D_; d += blockDim.x) {
        float s = 0.f;
#pragma unroll
        for (int w = 0; w < W_; ++w) s += base[(size_t)w * D_ + d];
        out[d] = s * (1.f / W_);
    }
}

// ---------------- K1b: per-entity attention (span mean-pool of rows) ----------------
__global__ void k_ent_att(const float* __restrict__ att,
                          const int* __restrict__ starts,
                          float* __restrict__ ws) {
    int beh = blockIdx.x;                // [0, B*E*H)
    int h   = beh % H_;
    int be  = beh / H_;
    int b   = be / E_;
    int st  = starts[be];
    const float* base = att + (((size_t)b * H_ + h) * L_ + st) * L_;
    float* out = ws + OFF_ENT_ATT + (size_t)beh * L_;   // layout [(b*E+e)*H+h][L]
    for (int l = threadIdx.x; l < L_; l += blockDim.x) {
        float s = 0.f;
#pragma unroll
        for (int w = 0; w < W_; ++w) s += base[(size_t)w * L_ + l];
        out[l] = s * (1.f / W_);
    }
}

// ---------------- K2: pair attention product, head-sum, normalize ----------------
__global__ void k_pair_attn(float* __restrict__ ws) {
    int bp = blockIdx.x;                 // [0, B*P)
    int b = bp / P_, p = bp % P_;
    int s = p / (E_ - 1);
    int r = p % (E_ - 1);
    int o = (r < s) ? r : r + 1;
    const float* ea_s = ws + OFF_ENT_ATT + ((size_t)(b * E_ + s) * H_) * L_;
    const float* ea_o = ws + OFF_ENT_ATT + ((size_t)(b * E_ + o) * H_) * L_;
    float* aout = ws + OFF_A + ((size_t)b * PPAD + p) * L_;

    constexpr int PER = L_ / 256;        // 8
    float qv[PER];
    float lsum = 0.f;
    int t = threadIdx.x;
#pragma unroll
    for (int i = 0; i < PER; ++i) {
        int l = t + i * 256;
        float q = 0.f;
#pragma unroll
        for (int h = 0; h < H_; ++h)
            q += ea_s[(size_t)h * L_ + l] * ea_o[(size_t)h * L_ + l];
        qv[i] = q;
        lsum += q;
    }
    __shared__ float red[256];
    red[t] = lsum;
    __syncthreads();
    for (int off = 128; off > 0; off >>= 1) {
        if (t < off) red[t] += red[t + off];
        __syncthreads();
    }
    float inv = 1.f / red[0];
#pragma unroll
    for (int i = 0; i < PER; ++i) aout[t + i * 256] = qv[i] * inv;
}

// ---------------- K3: c = a @ seq  (f32 WMMA, 2Mx2N register tiling) ----------------
// Per batch: [256 x 2048] @ [2048 x 768]. Each wave computes a 32x32 output block
// (2x2 grid of 16x16 WMMA tiles): 4 WMMAs per K-step of 4, reusing both A
// fragments and both B fragments -> 8 bytes/lane loaded per WMMA.
__global__ void k_context_wmma(const float* __restrict__ seq, float* __restrict__ ws) {
    constexpr int MT2 = PPAD / 32;       // 8  (32-row blocks)
    constexpr int NT2 = D_ / 32;         // 24 (32-col blocks)
    int id = blockIdx.x;
    int nt = id % NT2; id /= NT2;
    int mt = id % MT2;
    int b  = id / MT2;

    int lane = threadIdx.x;              // wave32: one wave per block
    int half = lane >> 4;                // 0 -> K{0,1}, 1 -> K{2,3}
    int l16  = lane & 15;
    int m0 = mt * 32 + l16;              // A rows for the two M tiles
    int m1 = m0 + 16;
    int n0 = nt * 32 + l16;              // B cols for the two N tiles
    int n1 = n0 + 16;

    const float* arow0 = ws + OFF_A + (size_t)b * PPAD * L_ + (size_t)m0 * L_;
    const float* arow1 = arow0 + (size_t)16 * L_;
    const float* Sq    = seq + (size_t)b * L_ * D_;

    v8f acc00 = {}, acc01 = {}, acc10 = {}, acc11 = {};
#pragma unroll 2
    for (int k0 = 0; k0 < L_; k0 += 4) {
        int k = k0 + half * 2;
        v2f av0, av1, bv0, bv1;
        av0.x = arow0[k];                av0.y = arow0[k + 1];
        av1.x = arow1[k];                av1.y = arow1[k + 1];
        bv0.x = Sq[(size_t)k * D_ + n0];       bv0.y = Sq[(size_t)(k + 1) * D_ + n0];
        bv1.x = Sq[(size_t)k * D_ + n1];       bv1.y = Sq[(size_t)(k + 1) * D_ + n1];
        acc00 = __builtin_amdgcn_wmma_f32_16x16x4_f32(false, av0, false, bv0, (short)0, acc00, false, false);
        acc01 = __builtin_amdgcn_wmma_f32_16x16x4_f32(false, av0, false, bv1, (short)0, acc01, false, false);
        acc10 = __builtin_amdgcn_wmma_f32_16x16x4_f32(false, av1, false, bv0, (short)0, acc10, false, false);
        acc11 = __builtin_amdgcn_wmma_f32_16x16x4_f32(false, av1, false, bv1, (short)0, acc11, false, false);
    }

    // C/D layout: VGPR j -> M = tile_base + j + 8*half, N = lane&15 within tile
    float* C = ws + OFF_C + (size_t)b * P_ * D_;
#pragma unroll
    for (int j = 0; j < 8; ++j) {
        int ma = mt * 32 + j + half * 8;       // M-tile 0
        int mb = ma + 16;                      // M-tile 1
        if (ma < P_) {
            C[(size_t)ma * D_ + n0] = acc00[j];
            C[(size_t)ma * D_ + n1] = acc01[j];
        }
        if (mb < P_) {
            C[(size_t)mb * D_ + n0] = acc10[j];
            C[(size_t)mb * D_ + n1] = acc11[j];
        }
    }
}

// ---------------- K4: assemble embs = [e_s | e_o | c] ----------------
__global__ void k_build_embs(float* __restrict__ ws) {
    int bp = blockIdx.x;                 // [0, B*P)
    int b = bp / P_, p = bp % P_;
    int s = p / (E_ - 1);
    int r = p % (E_ - 1);
    int o = (r < s) ? r : r + 1;
    const float* es = ws + OFF_ENT_EMB + (size_t)(b * E_ + s) * D_;
    const float* eo = ws + OFF_ENT_EMB + (size_t)(b * E_ + o) * D_;
    const float* c  = ws + OFF_C + ((size_t)b * P_ + p) * D_;
    float* out = ws + OFF_EMBS + ((size_t)b * PPAD + p) * K3D;
    for (int d = threadIdx.x; d < D_; d += blockDim.x) {
        out[d]           = es[d];
        out[D_ + d]      = eo[d];
        out[2 * D_ + d]  = c[d];
    }
}

// ---------------- K5: scores = embs @ [rel;nota]^T  (f32 WMMA) ----------------
__global__ void k_score_wmma(const float* __restrict__ rel,
                             const float* __restrict__ nota,
                             float* __restrict__ ws) {
    constexpr int MT = PPAD / 16;        // 16
    constexpr int NT = CB_ / 16;         // 5
    int id = blockIdx.x;
    int nt = id % NT; id /= NT;
    int mt = id % MT;
    int b  = id / MT;

    int lane = threadIdx.x;
    int half = lane >> 4;
    int l16  = lane & 15;
    int m_row = mt * 16 + l16;
    int n_col = nt * 16 + l16;

    const float* Em = ws + OFF_EMBS + (size_t)b * PPAD * K3D + (size_t)m_row * K3D;
    const float* crow;
    bool cvalid = true;
    if (n_col < R_)            crow = rel  + (size_t)n_col * K3D;
    else if (n_col < R_ + NN_) crow = nota + (size_t)(n_col - R_) * K3D;
    else { crow = rel; cvalid = false; }   // padded columns -> zeros

    v8f acc = {};
#pragma unroll 4
    for (int k0 = 0; k0 < K3D; k0 += 4) {
        int k = k0 + half * 2;
        v2f av;
        av.x = Em[k];
        av.y = Em[k + 1];
        v2f bv;
        bv.x = cvalid ? crow[k] : 0.f;
        bv.y = cvalid ? crow[k + 1] : 0.f;
        acc = __builtin_amdgcn_wmma_f32_16x16x4_f32(
            false, av, false, bv, (short)0, acc, false, false);
    }

    float* Sc = ws + OFF_SCORES + (size_t)b * PPAD * CB_;
#pragma unroll
    for (int j = 0; j < 8; ++j) {
        int m = mt * 16 + j + half * 8;
        Sc[(size_t)m * CB_ + n_col] = acc[j];   // padded rows written, never read
    }
}

// ---------------- K6: epilogue — nota max + concat ----------------
__global__ void k_epilogue(float* __restrict__ out, const float* __restrict__ ws) {
    int bp = blockIdx.x;                 // [0, B*P)
    int b = bp / P_, p = bp % P_;
    const float* sc = ws + OFF_SCORES + ((size_t)b * PPAD + p) * CB_;
    float* o = out + (size_t)bp * (R_ + 1);
    int t = threadIdx.x;
    if (t < R_) o[1 + t] = sc[t];
    if (t == R_) {
        float m = sc[R_];
#pragma unroll
        for (int i = 1; i < NN_; ++i) m = fmaxf(m, sc[R_ + i]);
        o[0] = m;
    }
}

// ---------------- launcher ----------------
extern "C" void kernel_launch(void* const* d_in, const int* in_sizes, int n_in,
                              void* d_out, int out_size, void* d_ws, size_t ws_size,
                              hipStream_t stream) {
    const float* seq    = (const float*)d_in[0];   // [B,L,D]
    const float* att    = (const float*)d_in[1];   // [B,H,L,L]
    const float* rel    = (const float*)d_in[2];   // [R,3D]
    const float* nota   = (const float*)d_in[3];   // [N,3D]
    const int*   starts = (const int*)  d_in[4];   // [B,E]
    float* out = (float*)d_out;
    float* ws  = (float*)d_ws;

    k_ent_emb   <<<B_ * E_,            256, 0, stream>>>(seq, starts, ws);
    k_ent_att   <<<B_ * E_ * H_,       256, 0, stream>>>(att, starts, ws);
    k_pair_attn <<<B_ * P_,            256, 0, stream>>>(ws);
    k_context_wmma<<<B_ * (PPAD/32) * (D_/32), 32, 0, stream>>>(seq, ws);
    k_build_embs<<<B_ * P_,            256, 0, stream>>>(ws);
    k_score_wmma<<<B_ * (PPAD/16) * (CB_/16), 32, 0, stream>>>(rel, nota, ws);
    k_epilogue  <<<B_ * P_,             64, 0, stream>>>(out, ws);
}